// OnlineNeuron_86904368267453
// MI455X (gfx1250) — compile-verified
//
#include <hip/hip_runtime.h>
#include <math.h>

// Online SNN neuron (T=4 recurrence), pure HBM-streaming kernel for MI455X.
// AI ~0.9 FLOP/byte -> memory bound; optimized for b128 NT loads/stores.

typedef float v4f __attribute__((ext_vector_type(4)));

#define T_STEPS 4
#define V_TH 1.0f

__device__ __forceinline__ float sigmoidf(float v) {
    return 1.0f / (1.0f + __expf(-v));
}

// Vectorized main kernel: each thread owns 4 contiguous neurons across all
// T=4 timesteps. Loads/stores are GLOBAL_LOAD_B128 / GLOBAL_STORE_B128 with
// non-temporal hints (stream-once data; 512MB working set >> 192MB L2).
__global__ void __launch_bounds__(256)
snn_online_vec4_kernel(const v4f* __restrict__ x, v4f* __restrict__ out,
                       const float* __restrict__ pa1, const float* __restrict__ pb1,
                       const float* __restrict__ pa2, const float* __restrict__ pb2,
                       unsigned int plane_vec)  // float4-elements per time plane
{
    unsigned int i = blockIdx.x * blockDim.x + threadIdx.x;
    if (i >= plane_vec) return;

    // Uniform scalar gates (s_load + a few VALU; negligible).
    const float a1 = sigmoidf(pa1[0]) - 0.5f;
    const float b1 = sigmoidf(pb1[0]) - 0.5f;
    const float a2 = sigmoidf(pa2[0]) + 0.5f;
    const float b2 = sigmoidf(pb2[0]) + 0.5f;

    v4f vd = {0.0f, 0.0f, 0.0f, 0.0f};          // v_d0 = 0
    v4f vs = {0.5f, 0.5f, 0.5f, 0.5f};          // v_s0 = 0.5 * V_TH

#pragma unroll
    for (int t = 0; t < T_STEPS; ++t) {
        // Independent of the recurrence chain -> compiler hoists all 4 loads.
        v4f xt = __builtin_nontemporal_load(&x[(size_t)t * plane_vec + i]);

        vd = a1 * vd + b1 * vs + xt;
        vs = a2 * vs + b2 * vd;

        v4f o;
#pragma unroll
        for (int k = 0; k < 4; ++k) {
            float u = vs[k] - V_TH;             // match reference boundary
            float s = (u >= 0.0f) ? V_TH : 0.0f;
            o[k] = s;
            vs[k] = vs[k] - s;                  // reset by detached spike
        }

        __builtin_nontemporal_store(o, &out[(size_t)t * plane_vec + i]);
    }
}

// Scalar tail for plane sizes not divisible by 4 (not hit for the reference
// shape, but keeps the kernel general).
__global__ void __launch_bounds__(256)
snn_online_tail_kernel(const float* __restrict__ x, float* __restrict__ out,
                       const float* __restrict__ pa1, const float* __restrict__ pb1,
                       const float* __restrict__ pa2, const float* __restrict__ pb2,
                       unsigned int tail_start, unsigned int plane)
{
    unsigned int i = tail_start + blockIdx.x * blockDim.x + threadIdx.x;
    if (i >= plane) return;

    const float a1 = sigmoidf(pa1[0]) - 0.5f;
    const float b1 = sigmoidf(pb1[0]) - 0.5f;
    const float a2 = sigmoidf(pa2[0]) + 0.5f;
    const float b2 = sigmoidf(pb2[0]) + 0.5f;

    float vd = 0.0f, vs = 0.5f;
#pragma unroll
    for (int t = 0; t < T_STEPS; ++t) {
        float xt = x[(size_t)t * plane + i];
        vd = a1 * vd + b1 * vs + xt;
        vs = a2 * vs + b2 * vd;
        float u = vs - V_TH;
        float s = (u >= 0.0f) ? V_TH : 0.0f;
        out[(size_t)t * plane + i] = s;
        vs -= s;
    }
}

extern "C" void kernel_launch(void* const* d_in, const int* in_sizes, int n_in,
                              void* d_out, int out_size, void* d_ws, size_t ws_size,
                              hipStream_t stream) {
    (void)n_in; (void)d_ws; (void)ws_size; (void)out_size;

    const float* x   = (const float*)d_in[0];
    const float* a1p = (const float*)d_in[1];
    const float* b1p = (const float*)d_in[2];
    const float* a2p = (const float*)d_in[3];
    const float* b2p = (const float*)d_in[4];
    float* out = (float*)d_out;

    const size_t total = (size_t)in_sizes[0];     // T * B * C * H * W
    const size_t plane = total / T_STEPS;         // neurons per timestep
    const unsigned int plane_vec = (unsigned int)(plane / 4);  // float4 groups
    const unsigned int tail_start = plane_vec * 4;
    const unsigned int tail = (unsigned int)(plane - tail_start);

    if (plane_vec > 0) {
        const int block = 256;
        const unsigned int grid = (plane_vec + block - 1) / block;
        snn_online_vec4_kernel<<<grid, block, 0, stream>>>(
            (const v4f*)x, (v4f*)out, a1p, b1p, a2p, b2p, plane_vec);
    }
    if (tail > 0) {
        const int block = 256;
        const unsigned int grid = (tail + block - 1) / block;
        snn_online_tail_kernel<<<grid, block, 0, stream>>>(
            x, out, a1p, b1p, a2p, b2p, tail_start, (unsigned int)plane);
    }
}